// Extrema1D_27504970563627
// MI455X (gfx1250) — compile-verified
//
#include <hip/hip_runtime.h>
#include <cstdint>

#define ROW_LEN   4096
#define THREADS   256
#define ELEMS     (ROW_LEN / THREADS)   // 16
#define MIN_DIST  32

// One workgroup per row. Greedy NMS == iterative argmax:
// the largest unsuppressed candidate is always the next primary.
__global__ __launch_bounds__(THREADS)
void Extrema1D_nms_kernel(const float* __restrict__ x, float* __restrict__ out)
{
    __shared__ __align__(16) float xs[ROW_LEN];      // 16 KB  row data
    __shared__ __align__(16) float outv[ROW_LEN];    // 16 KB  staged result
    __shared__ unsigned long long comp[ROW_LEN];     // 32 KB  packed (key, idx)
    __shared__ unsigned long long red[THREADS / 32]; // per-wave partial maxima
    __shared__ unsigned long long bcast;

    const int t   = threadIdx.x;
    const int row = blockIdx.x;
    const unsigned long long gp =
        (unsigned long long)(uintptr_t)(x + (size_t)row * ROW_LEN);

    // ---- CDNA5 async global->LDS copy of the row (ASYNCcnt-tracked) ----
    {
        // flat shared address truncated to 32 bits == LDS byte offset
        unsigned int xs_base = (unsigned int)(uintptr_t)(&xs[0]);
        #pragma unroll
        for (int m = 0; m < 4; ++m) {
            unsigned int voff  = (unsigned int)(t + THREADS * m) * 16u; // B128 chunks
            unsigned int laddr = xs_base + voff;
            asm volatile("global_load_async_to_lds_b128 %0, %1, %2"
                         :: "v"(laddr), "v"(voff), "s"(gp) : "memory");
        }
    }
    #pragma unroll
    for (int k = 0; k < ELEMS; ++k) outv[t + THREADS * k] = 0.0f;
    asm volatile("s_wait_asynccnt 0x0" ::: "memory");
    __syncthreads();

    // ---- extrema mask + packed key: high32 = bits(|x|)+1 (0 => invalid),
    //      low32 = (L-1-i) so larger composite == larger key, then smaller idx
    #pragma unroll
    for (int k = 0; k < ELEMS; ++k) {
        int   i = t + THREADS * k;
        float v = xs[i];
        bool right = (i < ROW_LEN - 1) && (xs[i + 1] > v); // pad(dx,(0,1)) > 0
        bool left  = (i == 0) || (v <= xs[i - 1]);         // pad(dx,(1,0)) <= 0
        bool neg   = (v <= 0.0f);
        bool valley =  right &&  left &&  neg;
        bool peak   = !right && !left && !neg;
        unsigned long long c = 0ull;
        if (valley || peak) {
            unsigned int kb = __float_as_uint(fabsf(v)); // >=0 floats: monotone bits
            c = (((unsigned long long)(kb + 1u)) << 32)
                | (unsigned int)(ROW_LEN - 1 - i);
        }
        comp[i] = c;
    }
    __syncthreads();

    const int lane = t & 31;
    const int wave = t >> 5;

    // ---- iterative argmax NMS (terminates in <= ~126 rounds) ----
    for (int it = 0; it < ROW_LEN; ++it) {
        unsigned long long local = 0ull;
        #pragma unroll
        for (int k = 0; k < ELEMS; ++k) {
            unsigned long long c = comp[t + THREADS * k];
            local = (c > local) ? c : local;
        }
        // wave32 butterfly reduction
        #pragma unroll
        for (int off = 16; off > 0; off >>= 1) {
            unsigned long long o = __shfl_xor(local, off, 32);
            local = (o > local) ? o : local;
        }
        if (lane == 0) red[wave] = local;
        __syncthreads();
        if (t == 0) {
            unsigned long long m = red[0];
            #pragma unroll
            for (int w = 1; w < THREADS / 32; ++w) {
                unsigned long long c = red[w];
                m = (c > m) ? c : m;
            }
            bcast = m;
        }
        __syncthreads();
        unsigned long long best = bcast;
        if ((best >> 32) == 0ull) break;   // no candidates left (uniform)

        int p = (ROW_LEN - 1) - (int)(best & 0xFFFFFFFFull);
        if (t == 0) outv[p] = xs[p];       // primary survives
        if (t < 2 * MIN_DIST + 1) {        // suppress +/-32 window
            int j = p - MIN_DIST + t;
            if (j >= 0 && j < ROW_LEN) comp[j] = 0ull;
        }
        __syncthreads();
    }

    // ---- write staged row out (B128 stores) ----
    #pragma unroll
    for (int m = 0; m < 4; ++m) {
        int c = t + THREADS * m;
        ((float4*)out)[(size_t)row * (ROW_LEN / 4) + c] =
            ((const float4*)outv)[c];
    }
}

extern "C" void kernel_launch(void* const* d_in, const int* in_sizes, int n_in,
                              void* d_out, int out_size, void* d_ws, size_t ws_size,
                              hipStream_t stream) {
    const float* x  = (const float*)d_in[0];
    float*       out = (float*)d_out;
    const int rows = in_sizes[0] / ROW_LEN;     // 128 * 1 * 4096 -> 128 rows
    Extrema1D_nms_kernel<<<rows, THREADS, 0, stream>>>(x, out);
}